// GlobalAttention_30288109371802
// MI455X (gfx1250) — compile-verified
//
#include <hip/hip_runtime.h>
#include <hip/hip_bf16.h>

typedef float v2f __attribute__((ext_vector_type(2)));
typedef float v8f __attribute__((ext_vector_type(8)));

#define BB 64
#define SS 2048
#define DD 512
#define MTILE 32            // s-rows per block (2 WMMA M-tiles)
#define KC 16               // K-chunk staged per double-buffer step
#define NCHUNK (DD / KC)    // 32
#define ASTR 516            // A slab LDS stride (512+4): conflict-free b64 reads
#define BSTR 20             // B chunk LDS stride (16+4): conflict-free b64 reads

// ---------------- Kernel 1: wq[b][e] = bq[e] + sum_d source[b][d] * Wq[e][d]
__global__ void ga_wq_kernel(const float* __restrict__ src,
                             const float* __restrict__ Wq,
                             const float* __restrict__ bq,
                             float* __restrict__ wq) {
  int b = blockIdx.x;
  int e = threadIdx.x;                 // 512 threads
  const float* srow = src + (size_t)b * DD;
  const float* wrow = Wq + (size_t)e * DD;
  float acc = bq[e];
  for (int d = 0; d < DD; ++d) acc += srow[d] * wrow[d];
  wq[(size_t)b * DD + e] = acc;
}

// ---------------- Kernel 2: scores[b][s] = sum_e v[e]*tanh(wq[b][e] + (mem[b] @ Wc^T)[s][e])
// Block = (32-row s-tile, batch b); 256 threads = 8 waves; wave w owns e-tiles [4w, 4w+4),
// and both M-tiles. Wc streamed through LDS in double-buffered K-chunks.
__global__ void __launch_bounds__(256) ga_scores_kernel(
    const float* __restrict__ mem,
    const float* __restrict__ Wc,
    const float* __restrict__ v,
    const float* __restrict__ wq,
    float* __restrict__ scores) {
  __shared__ float ldsA[MTILE * ASTR];        // 66048 B
  __shared__ float ldsB[2][DD * BSTR];        // 2 x 40960 B
  __shared__ float sScore[MTILE];

  const int tid = threadIdx.x;
  const int b  = blockIdx.y;
  const int s0 = blockIdx.x * MTILE;

  // ---- Stage A slab once: mem[b][s0..s0+31][0..511] -> LDS (coalesced float4)
  const float4* gA4 = (const float4*)(mem + ((size_t)b * SS + s0) * DD);
  for (int idx = tid; idx < MTILE * (DD / 4); idx += 256) {
    int row = idx >> 7;                       // DD/4 = 128 float4 per row
    int c4  = idx & 127;
    *(float4*)&ldsA[row * ASTR + c4 * 4] = gA4[(size_t)row * (DD / 4) + c4];
  }
  if (tid < MTILE) sScore[tid] = 0.0f;

  // ---- Stage B chunk 0: Wc[e=0..511][k=0..KC) -> ldsB[0]
  {
#pragma unroll
    for (int it = 0; it < 8; ++it) {
      int f = tid + it * 256;                 // 0..2047 float4 slots
      int e = f >> 2, j = f & 3;
      float4 val = *(const float4*)(Wc + (size_t)e * DD + j * 4);
      *(float4*)&ldsB[0][e * BSTR + j * 4] = val;
    }
  }
  __syncthreads();

  const int wave = tid >> 5;
  const int lane = tid & 31;
  const int half = lane >> 4;                 // 0: K0={0,1}, C rows 0-7 ; 1: K0={2,3}, rows 8-15
  const int ln   = lane & 15;                 // A: row M ; B/C/D: col N
  const int K0   = half * 2;

  int eCol[4];
#pragma unroll
  for (int i = 0; i < 4; ++i) eCol[i] = (wave * 4 + i) * 16 + ln;

  v8f acc[2][4] = {};                         // [m-tile][e-tile]

  for (int c = 0; c < NCHUNK; ++c) {
    // Prefetch next chunk into registers (loads fly during the WMMAs below)
    float4 pre[8];
    if (c + 1 < NCHUNK) {
#pragma unroll
      for (int it = 0; it < 8; ++it) {
        int f = tid + it * 256;
        int e = f >> 2, j = f & 3;
        pre[it] = *(const float4*)(Wc + (size_t)e * DD + (c + 1) * KC + j * 4);
      }
    }

    // Compute this chunk from LDS: KC/4 = 4 k-steps x 2 m-tiles x 4 e-tiles = 32 WMMAs
    const float* bufB = ldsB[c & 1];
    const int kcBase = c * KC;
#pragma unroll
    for (int kk = 0; kk < KC; kk += 4) {
      v2f aF[2];
#pragma unroll
      for (int m = 0; m < 2; ++m)
        aF[m] = *(const v2f*)&ldsA[(m * 16 + ln) * ASTR + kcBase + kk + K0];
#pragma unroll
      for (int i = 0; i < 4; ++i) {
        v2f bF = *(const v2f*)&bufB[eCol[i] * BSTR + kk + K0];
#pragma unroll
        for (int m = 0; m < 2; ++m) {
          acc[m][i] = __builtin_amdgcn_wmma_f32_16x16x4_f32(
              false, aF[m], false, bF, (short)0, acc[m][i], false, false);
        }
      }
    }

    // Store prefetched chunk (WAR on this buffer guarded by previous iteration's barrier)
    if (c + 1 < NCHUNK) {
      float* dst = ldsB[(c + 1) & 1];
#pragma unroll
      for (int it = 0; it < 8; ++it) {
        int f = tid + it * 256;
        int e = f >> 2, j = f & 3;
        *(float4*)&dst[e * BSTR + j * 4] = pre[it];
      }
    }
    __syncthreads();
  }

  // ---- Epilogue: p[m][r] = partial score of row m*16 + (half? r+8 : r)
  float p[2][8] = {};
#pragma unroll
  for (int i = 0; i < 4; ++i) {
    float wqv = wq[(size_t)b * DD + eCol[i]];
    float vv  = v[eCol[i]];
#pragma unroll
    for (int m = 0; m < 2; ++m)
#pragma unroll
      for (int r = 0; r < 8; ++r)
        p[m][r] += vv * tanhf(acc[m][i][r] + wqv);
  }

  // Reduce across the 16 lanes of each half (xor offsets < 16 stay in-half)
#pragma unroll
  for (int m = 0; m < 2; ++m)
#pragma unroll
    for (int r = 0; r < 8; ++r) {
      p[m][r] += __shfl_xor(p[m][r], 1, 32);
      p[m][r] += __shfl_xor(p[m][r], 2, 32);
      p[m][r] += __shfl_xor(p[m][r], 4, 32);
      p[m][r] += __shfl_xor(p[m][r], 8, 32);
    }
  if (ln == 0) {
#pragma unroll
    for (int m = 0; m < 2; ++m)
#pragma unroll
      for (int r = 0; r < 8; ++r)
        atomicAdd(&sScore[m * 16 + half * 8 + r], p[m][r]);  // ds_add_f32
  }
  __syncthreads();

  if (tid < MTILE) scores[(size_t)b * SS + s0 + tid] = sScore[tid];
}

// ---------------- Kernel 3: masked softmax over S; writes normalized align into d_out
__global__ void ga_softmax_kernel(const float* __restrict__ scores,
                                  const int* __restrict__ lens,
                                  float* __restrict__ out) {
  __shared__ float red[256];
  const int b = blockIdx.x;
  const int tid = threadIdx.x;
  int len = lens[b];
  if (len < 1) len = 1;
  if (len > SS) len = SS;

  const float* sc = scores + (size_t)b * SS;
  float* al = out + (size_t)BB * DD + (size_t)b * SS;

  float m = -3.4e38f;
  for (int s = tid; s < len; s += 256) m = fmaxf(m, sc[s]);
  red[tid] = m;
  __syncthreads();
  for (int off = 128; off; off >>= 1) {
    if (tid < off) red[tid] = fmaxf(red[tid], red[tid + off]);
    __syncthreads();
  }
  float mx = red[0];
  __syncthreads();

  float sum = 0.f;
  for (int s = tid; s < SS; s += 256) {
    float ev = (s < len) ? expf(sc[s] - mx) : 0.0f;
    al[s] = ev;
    sum += ev;
  }
  red[tid] = sum;
  __syncthreads();
  for (int off = 128; off; off >>= 1) {
    if (tid < off) red[tid] += red[tid + off];
    __syncthreads();
  }
  float inv = 1.0f / red[0];
  __syncthreads();
  for (int s = tid; s < SS; s += 256) al[s] *= inv;
}

// ---------------- Kernel 4: c[b][e] = sum_s align[b][s] * mem[b][s][e]
__global__ void ga_context_kernel(const float* __restrict__ mem,
                                  const int* __restrict__ lens,
                                  const float* __restrict__ outAlign,
                                  float* __restrict__ out) {
  __shared__ float s_al[256];
  const int tid = threadIdx.x;
  const int b = blockIdx.y;
  const int e = blockIdx.x * 256 + tid;
  int len = lens[b];
  if (len < 1) len = 1;
  if (len > SS) len = SS;

  const float* al = outAlign + (size_t)BB * DD + (size_t)b * SS;
  float c0 = 0.f;
  for (int sb = 0; sb < len; sb += 256) {
    __syncthreads();
    int sidx = sb + tid;
    s_al[tid] = (sidx < SS) ? al[sidx] : 0.0f;
    __syncthreads();
    int lim = len - sb;
    if (lim > 256) lim = 256;
    const float* rowBase = mem + ((size_t)b * SS + sb) * DD + e;
    for (int j = 0; j < lim; ++j) c0 += s_al[j] * rowBase[(size_t)j * DD];
  }
  out[(size_t)b * DD + e] = c0;
}

extern "C" void kernel_launch(void* const* d_in, const int* in_sizes, int n_in,
                              void* d_out, int out_size, void* d_ws, size_t ws_size,
                              hipStream_t stream) {
  const float* src  = (const float*)d_in[0];   // [B, D]
  const float* mem  = (const float*)d_in[1];   // [B, S, D]
  const int*   lens = (const int*)d_in[2];     // [B]
  const float* Wq   = (const float*)d_in[3];   // [D, D]
  const float* bq   = (const float*)d_in[4];   // [D]
  const float* Wc   = (const float*)d_in[5];   // [D, D]
  const float* v    = (const float*)d_in[6];   // [D]
  float* out = (float*)d_out;                  // c [B,D] then align [B,S]

  float* ws_wq     = (float*)d_ws;             // B*D floats
  float* ws_scores = ws_wq + (size_t)BB * DD;  // B*S floats

  ga_wq_kernel<<<BB, DD, 0, stream>>>(src, Wq, bq, ws_wq);
  ga_scores_kernel<<<dim3(SS / MTILE, BB), 256, 0, stream>>>(mem, Wc, v, ws_wq, ws_scores);
  ga_softmax_kernel<<<BB, 256, 0, stream>>>(ws_scores, lens, out);
  ga_context_kernel<<<dim3(DD / 256, BB), 256, 0, stream>>>(mem, lens, out, out);
}